// PSRoIPool_75419625718482
// MI455X (gfx1250) — compile-verified
//
#include <hip/hip_runtime.h>

// PS-RoI pooling for MI455X (gfx1250).
// N=4, C=21*7*7=1029, H=W=100, R=2048, PH=PW=G=7, OD=21, SCALE=1/16.
// L2-resident gather-reduce: features (164.6MB) fit in 192MB L2; kernel is
// latency-bound, so the design maximizes per-thread outstanding loads and
// keeps L2 clean (NT stores for output). WMMA is inapplicable (no operand
// reuse, ~12 adds per output); CDNA5 async global->LDS path stages the roi
// descriptors (ASYNCcnt + s_wait_asynccnt).

#define PHB 7
#define PWB 7
#define ODC 21
#define FH 100
#define FW 100
#define OUT_PER_ROI (ODC * PHB * PWB)   // 1029 (== channel count)

__global__ __launch_bounds__(256)
void PSRoIPool_75419625718482_kernel(const float* __restrict__ feat,
                                     const float* __restrict__ rois,
                                     float* __restrict__ out, int R)
{
    __shared__ float s_roi[8];
    __shared__ int   s_hs[PHB], s_he[PHB], s_ws[PWB], s_we[PWB];
    __shared__ int   s_b;

    const int r   = blockIdx.x;
    const int tid = threadIdx.x;

    // ---- Stage the 5-float roi descriptor into LDS (CDNA5 async path) ----
#if __has_builtin(__builtin_amdgcn_global_load_async_to_lds_b32)
    {
        typedef __attribute__((address_space(1))) int* g1i_t;
        typedef __attribute__((address_space(3))) int* l3i_t;
        if (tid < 5) {
            const float* g = rois + (size_t)r * 5 + tid;
            __builtin_amdgcn_global_load_async_to_lds_b32(
                (g1i_t)g, (l3i_t)&s_roi[tid], /*offset=*/0, /*cpol=*/0);
        }
#if __has_builtin(__builtin_amdgcn_s_wait_asynccnt)
        __builtin_amdgcn_s_wait_asynccnt(0);
#else
        asm volatile("s_wait_asynccnt 0" ::: "memory");
#endif
    }
#else
    if (tid < 5) s_roi[tid] = rois[(size_t)r * 5 + tid];
#endif
    __syncthreads();

    // ---- Bin boundaries (bit-exact vs reference: rintf == round-half-even,
    //      no FMA contraction feeding floor/ceil, IEEE divide) ----
    if (tid < PHB) {
        const float sw = rintf(s_roi[1]) * 0.0625f;           // *2^-4 is exact
        const float sh = rintf(s_roi[2]) * 0.0625f;
        const float ew = rintf(s_roi[3] + 1.0f) * 0.0625f;
        const float eh = rintf(s_roi[4] + 1.0f) * 0.0625f;
        const float roi_w = fmaxf(ew - sw, 0.1f);
        const float roi_h = fmaxf(eh - sh, 0.1f);
        const float bw = __fdiv_rn(roi_w, (float)PWB);
        const float bh = __fdiv_rn(roi_h, (float)PHB);
        const float p  = (float)tid;
        s_hs[tid] = (int)fminf(fmaxf(floorf(__fadd_rn(__fmul_rn(p,        bh), sh)), 0.0f), (float)FH);
        s_he[tid] = (int)fminf(fmaxf(ceilf (__fadd_rn(__fmul_rn(p + 1.0f, bh), sh)), 0.0f), (float)FH);
        s_ws[tid] = (int)fminf(fmaxf(floorf(__fadd_rn(__fmul_rn(p,        bw), sw)), 0.0f), (float)FW);
        s_we[tid] = (int)fminf(fmaxf(ceilf (__fadd_rn(__fmul_rn(p + 1.0f, bw), sw)), 0.0f), (float)FW);
    }
    if (tid == 0) s_b = (int)s_roi[0];
    __syncthreads();

    const int    b     = s_b;
    const float* fbase = feat + (size_t)b * OUT_PER_ROI * (FH * FW);
    float*       obase = out  + (size_t)r * OUT_PER_ROI;

    // One lane per output element; chan == i by construction.
    for (int i = tid; i < OUT_PER_ROI; i += 256) {
        const int ctop = i / 49;
        const int rem  = i - ctop * 49;       // ph*7 + pw
        const int ph   = rem / 7;
        const int pw   = rem - ph * 7;
        const int hs = s_hs[ph], he = s_he[ph];
        const int ws = s_ws[pw], we = s_we[pw];

        const float* cp = fbase + (size_t)i * (FH * FW);
        float acc = 0.0f;
        for (int y = hs; y < he; ++y) {
            const float* row = cp + y * FW;
            for (int x = ws; x < we; ++x)
                acc += row[x];
        }
        const int   area = (he - hs) * (we - ws);
        const float v    = (area > 0) ? __fdiv_rn(acc, (float)area) : 0.0f;
        // NT store: don't let the streamed output evict L2-resident features.
        __builtin_nontemporal_store(v, obase + i);
    }
}

extern "C" void kernel_launch(void* const* d_in, const int* in_sizes, int n_in,
                              void* d_out, int out_size, void* d_ws, size_t ws_size,
                              hipStream_t stream)
{
    const float* feat = (const float*)d_in[0];   // (4, 1029, 100, 100) f32
    const float* rois = (const float*)d_in[1];   // (2048, 5) f32
    float*       out  = (float*)d_out;           // (2048, 21, 7, 7) f32
    const int R = in_sizes[1] / 5;

    PSRoIPool_75419625718482_kernel<<<R, 256, 0, stream>>>(feat, rois, out, R);
}